// ROIPooling_63574105915534
// MI455X (gfx1250) — compile-verified
//
#include <hip/hip_runtime.h>
#include <float.h>
#include <stdint.h>

// ---------------------------------------------------------------------------
// ROI adaptive max pool (7x7) for f[256,64,64] fp32, 128 ROIs.
//
// MI455X strategy: this op is max-reduction (no matmul -> WMMA inapplicable).
// Feature map (4MB) is L2-resident; kernel is L2/LDS/VALU bound. We use the
// CDNA5 Tensor Data Mover to DMA each ROI's [Cchunk x roiH x roiW] tile into
// LDS as one packed transfer (avoids uncoalesced short-row per-lane loads and
// re-reads), then 8 waves compute 49 window maxes per channel from LDS.
// TDM issue is guarded by a *scalar* (readfirstlane) branch: TDM ignores
// EXEC, so an EXEC-masked guard alone could duplicate-issue from all waves.
// ---------------------------------------------------------------------------

typedef unsigned int u32x4 __attribute__((ext_vector_type(4)));
typedef int          i32x4 __attribute__((ext_vector_type(4)));
typedef int          i32x8 __attribute__((ext_vector_type(8)));

#define SBINS      7
#define C_TOTAL    256
#define H_DIM      64
#define W_DIM      64
#define BUF_FLOATS 8192   // 32 KB per buffer, x2 buffers = 64 KB LDS

__global__ __launch_bounds__(256) void roipool_tdm_kernel(
    const float* __restrict__ fmap,
    const int*   __restrict__ rois,
    float*       __restrict__ out)
{
    __shared__ float buf[2 * BUF_FLOATS];

    const int r  = blockIdx.x;
    const int x1 = rois[r * 4 + 0];   // row range  [x1, x2]  (bins over H)
    const int y1 = rois[r * 4 + 1];   // col range  [y1, y2]  (bins over W)
    const int x2 = rois[r * 4 + 2];
    const int y2 = rois[r * 4 + 3];

    const int Lh = x2 - x1 + 1;       // tile height (rows)
    const int Lw = y2 - y1 + 1;       // tile width  (cols)
    const int tileFloats = Lh * Lw;   // <= 4096

    // channels per chunk, sized so a chunk fits one 32KB LDS buffer (>= 2)
    int CC = BUF_FLOATS / tileFloats;
    if (CC > C_TOTAL) CC = C_TOTAL;
    const int nChunks = (C_TOTAL + CC - 1) / CC;

    // byte address of tile origin (c=0) in global memory
    const unsigned long long gbase =
        (unsigned long long)(uintptr_t)fmap +
        4ull * (unsigned)(x1 * W_DIM + y1);

    // LDS byte offset of our buffer: flat shared pointer's low 32 bits are
    // the LDS byte address (shared aperture keeps offset in addr[31:0]).
    const unsigned ldsBase = (unsigned)(uintptr_t)&buf[0];

    // Uniform scalar guard: forces s_cmp + s_cbranch (wave-level skip), so
    // exactly one wave per workgroup issues TDM ops (TDM ignores EXEC).
    const int waveId = __builtin_amdgcn_readfirstlane((int)(threadIdx.x >> 5));
    const bool issuer = (waveId == 0);

    // Build a 3D tensor descriptor (D#) and issue TENSOR_LOAD_TO_LDS.
    // dims: dim0 = W (stride 64 elements), dim1 = H (stride 4096 elements),
    // dim2 = C. Tile = [cc, Lh, Lw] starting at (c0, x1, y1); lands packed
    // in LDS as cc * Lh * Lw consecutive floats.
    auto issue = [&](int k) {
        const int c0 = k * CC;
        int cc = C_TOTAL - c0; if (cc > CC) cc = CC;
        const unsigned long long ga = gbase + 4ull * 4096ull * (unsigned)c0;
        const unsigned lds_addr = ldsBase + (unsigned)((k & 1) * BUF_FLOATS * 4);

        u32x4 g0;
        g0[0] = 1u;                                           // count=1 (valid), user mode
        g0[1] = lds_addr;                                     // lds_addr (bytes)
        g0[2] = (unsigned)(ga & 0xFFFFFFFFull);               // global_addr[31:0]
        g0[3] = (unsigned)((ga >> 32) & 0x1FFFFFFull)         // global_addr[56:32]
              | (2u << 30);                                   // type = 2 ("image")

        i32x8 g1;
        g1[0] = (int)(2u << 16);                              // data_size = 4B; no mask/pad/iter
        g1[1] = (int)((unsigned)(W_DIM & 0xFFFF) << 16);      // barrier_addr=0 | tensor_dim0[15:0]
        g1[2] = (int)(((unsigned)W_DIM >> 16)                 // tensor_dim0[31:16]
              | ((unsigned)(H_DIM & 0xFFFF) << 16));          // | tensor_dim1[15:0]
        g1[3] = (int)(((unsigned)H_DIM >> 16)                 // tensor_dim1[31:16]
              | ((unsigned)Lw << 16));                        // | tile_dim0 (row length)
        g1[4] = (int)((unsigned)Lh                            // tile_dim1 (rows)
              | ((unsigned)cc << 16));                        // | tile_dim2 (channels)
        g1[5] = W_DIM;                                        // tensor_dim0_stride[31:0] = 64
        g1[6] = (int)(((unsigned)(H_DIM * W_DIM) & 0xFFFFu) << 16); // stride0 hi=0 | stride1[15:0]
        g1[7] = (int)((unsigned)(H_DIM * W_DIM) >> 16);       // tensor_dim1_stride[47:16]

        i32x4 g2;
        g2[0] = C_TOTAL;                                      // tensor_dim2
        g2[1] = 0;                                            // tensor_dim3
        g2[2] = 0;                                            // tensor_dim2_stride lo (unused, 3D)
        g2[3] = 0;                                            // stride hi | tile_dim3=0
        i32x4 g3 = {0, 0, 0, 0};

        // clang-23 / therock-10.0 lane: 6-arg form
        // (u32x4 g0, i32x8 g1, i32x4 g2, i32x4 g3, i32x8 aux, i32 cpol)
        i32x8 aux = {0, 0, 0, 0, 0, 0, 0, 0};
        __builtin_amdgcn_tensor_load_to_lds(g0, g1, g2, g3, aux, 0);
    };

    if (issuer) issue(0);

    for (int k = 0; k < nChunks; ++k) {
        // wave 0: prefetch next chunk, then wait for chunk k (TDM is in-order)
        if (issuer) {
            if (k + 1 < nChunks) {
                issue(k + 1);
                __builtin_amdgcn_s_wait_tensorcnt(1);
            } else {
                __builtin_amdgcn_s_wait_tensorcnt(0);
            }
        }
        __syncthreads();   // chunk k's LDS tile now visible to all 8 waves

        const int c0 = k * CC;
        int cc = C_TOTAL - c0; if (cc > CC) cc = CC;
        const float* tile = &buf[(k & 1) * BUF_FLOATS];

        const int nOut = cc * (SBINS * SBINS);
        for (int t = threadIdx.x; t < nOut; t += 256) {
            const int c = t / (SBINS * SBINS);
            const int b = t - c * (SBINS * SBINS);
            const int i = b / SBINS;            // row bin
            const int j = b - i * SBINS;        // col bin
            // PyTorch adaptive windows (tile-relative): [floor(iL/7), ceil((i+1)L/7))
            const int hs = (i * Lh) / SBINS;
            const int he = ((i + 1) * Lh + (SBINS - 1)) / SBINS;
            const int ws = (j * Lw) / SBINS;
            const int we = ((j + 1) * Lw + (SBINS - 1)) / SBINS;

            const float* cbase = tile + c * tileFloats;
            float m = -FLT_MAX;
            for (int h = hs; h < he; ++h) {
                const float* row = cbase + h * Lw;
                for (int w = ws; w < we; ++w) m = fmaxf(m, row[w]);
            }
            out[((size_t)r * C_TOTAL + (size_t)(c0 + c)) * (SBINS * SBINS) + b] = m;
        }
        __syncthreads();   // protect buf[k&1] before chunk k+2 DMA overwrites it
    }
}

extern "C" void kernel_launch(void* const* d_in, const int* in_sizes, int n_in,
                              void* d_out, int out_size, void* d_ws, size_t ws_size,
                              hipStream_t stream) {
    const float* fmap = (const float*)d_in[0];   // (1,256,64,64) fp32
    const int*   rois = (const int*)d_in[1];     // (1,128,4) int32 after stacking
    float*       out  = (float*)d_out;           // (1,128,256,49) fp32
    const int R = in_sizes[1] / 4;               // 128 ROIs
    roipool_tdm_kernel<<<R, 256, 0, stream>>>(fmap, rois, out);
}